// Ensemble_of_ensemble_45329084842599
// MI455X (gfx1250) — compile-verified
//
#include <hip/hip_runtime.h>
#include <math.h>

// ---------------------------------------------------------------------------
// Ensemble-of-ensembles forward, MI455X (gfx1250, wave32).
// One 256-thread block (8 waves) per batch row:
//   stage 16x345 logits row -> LDS (async-to-LDS), per-model softmax stats,
//   16-long head math on thread 0, both einsums via v_wmma_f32_16x16x4_f32,
//   per-row scalar partials -> d_ws, then a 1-block deterministic reduction.
// ---------------------------------------------------------------------------

typedef __attribute__((ext_vector_type(2))) float v2f;
typedef __attribute__((ext_vector_type(8))) float v8f;

#define B_TOT   8192
#define M_NUM   16
#define C_NUM   345
#define STRIDE  360            // padded LDS row stride: 2*360 % 64 == 16 -> conflict-free half-wave B loads
#define NTILE   22             // ceil(345/16)

#define O_POST   0                       // ems_out_post  : 8192*345
#define O_CHILD  2826240                 // child_loss    : 1
#define O_CONF   2826241                 // confidence    : 1
#define O_ENS    2826242                 // ensemble_loss : 1
#define O_WSOFT  2826243                 // wm_soft       : 8192*16
#define O_TCONF  2957315                 // true_confs    : 8192*16

#if __has_builtin(__builtin_amdgcn_global_load_async_to_lds_b32)
#define ASYNC_LDS 1
typedef __attribute__((address_space(1))) int* as1_i32p;
typedef __attribute__((address_space(3))) int* as3_i32p;
#endif

#if __has_builtin(__builtin_amdgcn_wmma_f32_16x16x4_f32)
#define HAVE_WMMA_F32 1
#endif

__global__ __launch_bounds__(256)
void ens_row_kernel(const float* __restrict__ yp, const int* __restrict__ labels,
                    float* __restrict__ out, float* __restrict__ ws)
{
    __shared__ float sL[M_NUM * STRIDE];
    __shared__ float s_tp[16], s_epl[16], s_tc[16], s_wm[16], s_pw[16], s_x[16], s_t2[16];
    __shared__ float s_emax[8], s_esum[8], s_eval[8];
    __shared__ float s_child, s_conf;

    const int tid   = threadIdx.x;
    const int wave  = tid >> 5;
    const int lane  = tid & 31;
    const int b     = blockIdx.x;
    const int label = labels[b];
    const float* gbase = yp + (size_t)b * (M_NUM * C_NUM);

    // ---- stage logits row into LDS; zero-pad cols [345,360) ----
    for (int idx = tid; idx < M_NUM * (STRIDE - C_NUM); idx += 256) {
        int m = idx / (STRIDE - C_NUM);
        int c = C_NUM + idx % (STRIDE - C_NUM);
        sL[m * STRIDE + c] = 0.f;
    }
#ifdef ASYNC_LDS
    for (int idx = tid; idx < M_NUM * C_NUM; idx += 256) {
        int m = idx / C_NUM, c = idx - m * C_NUM;
        __builtin_amdgcn_global_load_async_to_lds_b32(
            (as1_i32p)(float*)(gbase + idx),
            (as3_i32p)(&sL[m * STRIDE + c]),
            0, 0);
    }
#if __has_builtin(__builtin_amdgcn_s_wait_asynccnt)
    __builtin_amdgcn_s_wait_asynccnt(0);
#else
    asm volatile("s_wait_asynccnt 0" ::: "memory");
#endif
#else
    for (int idx = tid; idx < M_NUM * C_NUM; idx += 256) {
        int m = idx / C_NUM, c = idx - m * C_NUM;
        sL[m * STRIDE + c] = gbase[idx];
    }
#endif
    __syncthreads();

    // ---- per-model softmax stats: wave w handles models 2w, 2w+1 ----
    for (int i = 0; i < 2; ++i) {
        int m = 2 * wave + i;
        const float* Lr = &sL[m * STRIDE];
        float mx = -INFINITY;
        for (int c = lane; c < C_NUM; c += 32) mx = fmaxf(mx, Lr[c]);
        for (int off = 16; off; off >>= 1) mx = fmaxf(mx, __shfl_xor(mx, off, 32));
        float se = 0.f;
        for (int c = lane; c < C_NUM; c += 32) se += expf(Lr[c] - mx);
        for (int off = 16; off; off >>= 1) se += __shfl_xor(se, off, 32);
        if (lane == 0) {
            float Ll = Lr[label];
            s_tp[m]  = expf(Ll - mx) / se;        // probs[b, m, label]
            s_epl[m] = (mx + logf(se)) - Ll;      // -log_softmax at label
        }
    }
    __syncthreads();

    // ---- 16-long head math (thread 0; negligible vs 22KB row load) ----
    if (tid == 0) {
        float tmax = -INFINITY;
        for (int m = 0; m < 16; ++m) tmax = fmaxf(tmax, s_tp[m]);
        float ts = 0.f;
        for (int m = 0; m < 16; ++m) { float e = expf(s_tp[m] - tmax); s_tc[m] = e; ts += e; }
        for (int m = 0; m < 16; ++m) s_tc[m] /= ts;                 // true_confs = weights
        float wsum = 0.f;
        for (int m = 0; m < 16; ++m) wsum += fabsf(s_tc[m]);
        float wden = fmaxf(wsum, 1e-12f);
        for (int m = 0; m < 16; ++m) s_wm[m] = s_tc[m] / wden;      // weighted_mat
        // top-4 -> post_weights
        unsigned used = 0; float topsum = 0.f;
        for (int k = 0; k < 4; ++k) {
            int bi = 0; float bv = -INFINITY;
            for (int m = 0; m < 16; ++m)
                if (!((used >> m) & 1u) && s_tc[m] > bv) { bv = s_tc[m]; bi = m; }
            used |= 1u << bi; topsum += bv;
        }
        float pden = fmaxf(topsum, 1e-12f);
        for (int m = 0; m < 16; ++m) s_pw[m] = ((used >> m) & 1u) ? s_tc[m] / pden : 0.f;
        // x = softmax(weighted_mat) (= wm_soft), t2 = softmax(true_confs)
        float xm = -INFINITY, tm = -INFINITY;
        for (int m = 0; m < 16; ++m) { xm = fmaxf(xm, s_wm[m]); tm = fmaxf(tm, s_tc[m]); }
        float xs = 0.f, ts2 = 0.f;
        for (int m = 0; m < 16; ++m) {
            s_x[m]  = expf(s_wm[m] - xm); xs  += s_x[m];
            s_t2[m] = expf(s_tc[m] - tm); ts2 += s_t2[m];
        }
        float conf = 0.f, child = 0.f;
        for (int m = 0; m < 16; ++m) {
            s_x[m] /= xs; s_t2[m] /= ts2;
            float xv = s_x[m], tv = s_t2[m];
            conf  += fmaxf(xv, 0.f) - xv * tv + log1pf(expf(-fabsf(xv)));
            child += s_epl[m] * xv;
            out[O_WSOFT + (size_t)b * 16 + m] = xv;
            out[O_TCONF + (size_t)b * 16 + m] = s_tc[m];
        }
        s_child = child; s_conf = conf;
    }
    __syncthreads();

    // ---- einsums: D = A(16x4) x B(4x16) + C, A row0 = wm, row1 = pw ----
    const int half = lane >> 4;   // K half select
    const int ll   = lane & 15;   // M (for A) / N (for B,D)
    float lmax = -INFINITY, lsum = 0.f, lval = 0.f;

#ifdef HAVE_WMMA_F32
    v2f a[4];
    for (int j = 0; j < 4; ++j) {
        int k0 = 4 * j + 2 * half;
        float ax = 0.f, ay = 0.f;
        if (ll == 0)      { ax = s_wm[k0]; ay = s_wm[k0 + 1]; }
        else if (ll == 1) { ax = s_pw[k0]; ay = s_pw[k0 + 1]; }
        a[j].x = ax; a[j].y = ay;
    }
#endif

    for (int t = wave; t < NTILE; t += 8) {          // wave-uniform loop: EXEC full at WMMA
        int cc = 16 * t + ll;
        float ems, emsp;
#ifdef HAVE_WMMA_F32
        v8f acc = {0.f, 0.f, 0.f, 0.f, 0.f, 0.f, 0.f, 0.f};
        for (int j = 0; j < 4; ++j) {
            int r = 4 * j + 2 * half;
            v2f bv;
            bv.x = sL[r * STRIDE + cc];
            bv.y = sL[(r + 1) * STRIDE + cc];
            acc = __builtin_amdgcn_wmma_f32_16x16x4_f32(
                      false, a[j], false, bv, (short)0, acc, false, false);
        }
        ems = acc[0]; emsp = acc[1];                 // D rows M=0 / M=1 (lanes 0-15)
#else
        float e0 = 0.f, e1 = 0.f;
        for (int m = 0; m < 16; ++m) {
            float lv = sL[m * STRIDE + 16 * t + ll];
            e0 += s_wm[m] * lv; e1 += s_pw[m] * lv;
        }
        ems = e0; emsp = e1;
#endif
        if (half == 0 && cc < C_NUM) {
            out[O_POST + (size_t)b * C_NUM + cc] = emsp;
            float nm = fmaxf(lmax, ems);
            lsum = lsum * expf(lmax - nm) + expf(ems - nm);   // online logsumexp of ems_out
            lmax = nm;
            if (cc == label) lval = ems;
        }
    }

    // cross-lane logsumexp reduce (offset 16 first so invalid upper half merges cleanly)
    for (int off = 16; off; off >>= 1) {
        float om = __shfl_xor(lmax, off, 32);
        float os = __shfl_xor(lsum, off, 32);
        float nm = fmaxf(lmax, om);
        lsum = lsum * expf(lmax - nm) + os * expf(om - nm);
        lmax = nm;
        lval += __shfl_xor(lval, off, 32);
    }
    if (lane == 0) { s_emax[wave] = lmax; s_esum[wave] = lsum; s_eval[wave] = lval; }
    __syncthreads();

    if (tid == 0) {
        float gm = -INFINITY;
        for (int w = 0; w < 8; ++w) gm = fmaxf(gm, s_emax[w]);
        float gs = 0.f, gv = 0.f;
        for (int w = 0; w < 8; ++w) { gs += s_esum[w] * expf(s_emax[w] - gm); gv += s_eval[w]; }
        float ens = (gm + logf(gs)) - gv;            // -log_softmax(ems_out)[label]
        ws[(size_t)b * 3 + 0] = s_child;
        ws[(size_t)b * 3 + 1] = s_conf;
        ws[(size_t)b * 3 + 2] = ens;
    }
}

__global__ __launch_bounds__(256)
void ens_reduce_kernel(const float* __restrict__ ws, float* __restrict__ out)
{
    __shared__ float sc[256], sf[256], se[256];
    const int tid = threadIdx.x;
    float c = 0.f, f = 0.f, e = 0.f;
    for (int i = tid; i < B_TOT; i += 256) {
        c += ws[(size_t)i * 3 + 0];
        f += ws[(size_t)i * 3 + 1];
        e += ws[(size_t)i * 3 + 2];
    }
    sc[tid] = c; sf[tid] = f; se[tid] = e;
    __syncthreads();
    for (int s = 128; s; s >>= 1) {
        if (tid < s) { sc[tid] += sc[tid + s]; sf[tid] += sf[tid + s]; se[tid] += se[tid + s]; }
        __syncthreads();
    }
    if (tid == 0) {
        out[O_CHILD] = sc[0] / (float)(B_TOT * M_NUM);
        out[O_CONF]  = sf[0] / (float)(B_TOT * M_NUM);
        out[O_ENS]   = se[0] / (float)B_TOT;
    }
}

extern "C" void kernel_launch(void* const* d_in, const int* in_sizes, int n_in,
                              void* d_out, int out_size, void* d_ws, size_t ws_size,
                              hipStream_t stream)
{
    (void)in_sizes; (void)n_in; (void)out_size; (void)ws_size;
    const float* yp     = (const float*)d_in[0];
    const int*   labels = (const int*)d_in[1];
    float* out = (float*)d_out;
    float* ws  = (float*)d_ws;   // needs 8192*3 floats = 96 KB

    ens_row_kernel<<<B_TOT, 256, 0, stream>>>(yp, labels, out, ws);
    ens_reduce_kernel<<<1, 256, 0, stream>>>(ws, out);
}